// PostProcessor_47072841564452
// MI455X (gfx1250) — compile-verified
//
#include <hip/hip_runtime.h>
#include <hip/hip_bf16.h>
#include <math.h>

// ---------------------------------------------------------------------------
// Mask R-CNN box-head post-processor for MI455X (gfx1250, wave32).
//   B=8 images, N=1000 proposals, C=81 classes.
// Outputs (concatenated flat, float): dets [8,100,5] then labels [8,100]
// (labels stored as float values: 1..80 or -1).
// ---------------------------------------------------------------------------

#define BB   8
#define NN   1000
#define CC   81
#define NCLS 80                     // foreground classes
#define IMG_W 1333.0f
#define IMG_H 800.0f
#define SCORE_THRESH 0.05f
#define NMS_THRESH   0.5f
#define DETS_PER_IMG 100
#define DW_CLIP 4.135166556742356f  // log(1000/16)

typedef __attribute__((ext_vector_type(2))) float v2f;
typedef __attribute__((ext_vector_type(8))) float v8f;

// ---------------------------------------------------------------------------
// Kernel 1: softmax over 81 logits per row; writes prob of classes 1..80
// into scores[(b*80 + c-1)*N + n].  One wave handles 16 rows; the exp-row-sum
// is computed exactly in f32 with V_WMMA_F32_16X16X4_F32 against an all-ones
// B matrix (D[m, n] = sum_k A[m,k]).
// ---------------------------------------------------------------------------
__global__ __launch_bounds__(128)
void softmax_kernel(const float* __restrict__ logits, float* __restrict__ scores)
{
    const int wave = threadIdx.x >> 5;
    const int lane = threadIdx.x & 31;
    const int half = lane >> 4;          // 0: K pair {0,1}, 1: K pair {2,3}
    const int lrow = lane & 15;          // row within 16-row tile
    const int row0 = blockIdx.x * 64 + wave * 16;   // 8000 rows = 125 blocks * 64
    if (row0 >= BB * NN) return;         // wave-uniform guard (never taken here)
    const int row = row0 + lrow;
    const float* lg = logits + (size_t)row * CC;

    // per-row max (lanes l and l+16 duplicate the same row's work)
    float m = -1e30f;
    for (int c = 0; c < CC; ++c) m = fmaxf(m, lg[c]);

    float sum;
#if __has_builtin(__builtin_amdgcn_wmma_f32_16x16x4_f32)
    // rowsum via WMMA: A = 16x4 tile of exp(logit-max), B = ones(4x16).
    // A layout (32-bit, 16x4): lanes 0-15 carry K={0,1}, lanes 16-31 K={2,3}.
    v8f acc = {};
    const v2f ones = {1.0f, 1.0f};
    for (int k0 = 0; k0 < 84; k0 += 4) {
        const int c0 = k0 + half * 2;
        float e0 = (c0     < CC) ? __expf(lg[c0]     - m) : 0.0f;
        float e1 = (c0 + 1 < CC) ? __expf(lg[c0 + 1] - m) : 0.0f;
        v2f a = {e0, e1};
        acc = __builtin_amdgcn_wmma_f32_16x16x4_f32(
            /*neg_a=*/false, a, /*neg_b=*/false, ones,
            /*c_mod=*/(short)0, acc, /*reuse_a=*/false, /*reuse_b=*/false);
    }
    // D layout: VGPR v, lanes 0-15 -> M=v ; lanes 16-31 -> M=v+8.
    // rowsum[r] lives at lane r (r<8, elem r) or lane r+16 (r>=8, elem r-8).
    const int  sel     = lane & 7;
    const float partial = acc[sel];
    const int  src     = (lrow < 8) ? lrow : (lrow + 16);
    sum = __shfl(partial, src, 32);
#else
    sum = 0.0f;
    for (int c = 0; c < CC; ++c) sum += __expf(lg[c] - m);
#endif

    const float inv = 1.0f / sum;
    const int b = row / NN, n = row % NN;
    // lanes 0-15 write classes 1..40 of their row, lanes 16-31 classes 41..80
    const int cstart = 1 + half * 40;
    for (int t = 0; t < 40; ++t) {
        const int c = cstart + t;
        const float p = __expf(lg[c] - m) * inv;
        scores[((size_t)(b * NCLS) + (c - 1)) * NN + n] = p;
    }
}

// ---------------------------------------------------------------------------
// Kernel 2: BoxCoder.decode (TO_REMOVE=1) + clip_to_image for classes 1..80.
// boxes[((b*80+cc)*N + n)*4 + k], xyxy.
// ---------------------------------------------------------------------------
__global__ __launch_bounds__(256)
void decode_kernel(const float* __restrict__ reg, const float* __restrict__ props,
                   float* __restrict__ boxes)
{
    const int idx = blockIdx.x * blockDim.x + threadIdx.x;
    if (idx >= BB * NN * NCLS) return;
    const int cc  = idx % NCLS;          // 0..79 -> class cc+1
    const int row = idx / NCLS;          // 0..7999
    const int b = row / NN, n = row % NN;

    const float x1 = props[row * 4 + 0], y1 = props[row * 4 + 1];
    const float x2 = props[row * 4 + 2], y2 = props[row * 4 + 3];
    const float w  = x2 - x1 + 1.0f, h = y2 - y1 + 1.0f;
    const float cx = x1 + 0.5f * w,  cy = y1 + 0.5f * h;

    const float* r = reg + (size_t)row * (CC * 4) + (size_t)(cc + 1) * 4;
    const float dx = r[0] * 0.1f;                    // / WX=10
    const float dy = r[1] * 0.1f;
    const float dw = fminf(r[2] * 0.2f, DW_CLIP);    // / WW=5
    const float dh = fminf(r[3] * 0.2f, DW_CLIP);

    const float pcx = dx * w + cx, pcy = dy * h + cy;
    const float pw  = __expf(dw) * w, ph = __expf(dh) * h;

    float ox1 = pcx - 0.5f * pw, oy1 = pcy - 0.5f * ph;
    float ox2 = pcx + 0.5f * pw - 1.0f, oy2 = pcy + 0.5f * ph - 1.0f;
    ox1 = fminf(fmaxf(ox1, 0.0f), IMG_W - 1.0f);
    ox2 = fminf(fmaxf(ox2, 0.0f), IMG_W - 1.0f);
    oy1 = fminf(fmaxf(oy1, 0.0f), IMG_H - 1.0f);
    oy2 = fminf(fmaxf(oy2, 0.0f), IMG_H - 1.0f);

    float* o = boxes + (((size_t)(b * NCLS + cc)) * NN + n) * 4;
    o[0] = ox1; o[1] = oy1; o[2] = ox2; o[3] = oy2;
}

// ---------------------------------------------------------------------------
// Kernel 3: per-(image,class) greedy NMS.  One 256-thread workgroup per
// (b,c).  Bitonic sort of (score, idx) in LDS (descending score, ascending
// idx on ties = stable argsort(-scores)), then CDNA5 async global->LDS gather
// of sorted boxes, then greedy suppression, then scatter keep-masked scores
// back in original order (zeros for dropped -> buffer fully rewritten every
// launch, so replays are deterministic).
// ---------------------------------------------------------------------------
__device__ __forceinline__ bool before_ord(float ka, int ia, float kb, int ib) {
    return (ka > kb) || (ka == kb && ia < ib);
}

__global__ __launch_bounds__(256)
void nms_kernel(const float* __restrict__ scores, const float* __restrict__ boxes,
                float* __restrict__ kept_scores)
{
    __shared__ float s_key[1024];
    __shared__ int   s_idx[1024];
    __shared__ float s_box[1024][4];
    __shared__ unsigned char s_sup[1024];
    __shared__ int   s_nvalid;

    const int tid = threadIdx.x;
    const int pair = blockIdx.x;               // 0..639
    const int b = pair / NCLS, cc = pair % NCLS;
    const float* sc = scores + (size_t)pair * NN;
    const float* bx = boxes  + (size_t)pair * NN * 4;

    for (int i = tid; i < 1024; i += 256) {
        s_key[i] = (i < NN) ? sc[i] : -1e30f;
        s_idx[i] = i;
        s_sup[i] = 0;
    }
    __syncthreads();

    // bitonic sort, 1024 elements
    for (int k = 2; k <= 1024; k <<= 1) {
        for (int j = k >> 1; j > 0; j >>= 1) {
            for (int t = tid; t < 1024; t += 256) {
                const int x = t ^ j;
                if (x > t) {
                    const bool up = ((t & k) == 0);
                    const float kt = s_key[t], kx = s_key[x];
                    const int   it = s_idx[t], ix = s_idx[x];
                    const bool sw = up ? before_ord(kx, ix, kt, it)
                                       : before_ord(kt, it, kx, ix);
                    if (sw) { s_key[t] = kx; s_idx[t] = it ? 0 : 0, s_idx[t] = ix;
                              s_key[x] = kt; s_idx[x] = it; }
                }
            }
            __syncthreads();
        }
    }

    if (tid == 0) {
        int nv = 0;
        while (nv < NN && s_key[nv] > SCORE_THRESH) ++nv;
        s_nvalid = nv;
    }
    __syncthreads();
    const int nvalid = s_nvalid;

    // Async gather of boxes in sorted order into LDS (CDNA5 async-tensor path).
    for (int i = tid; i < NN; i += 256) {
        const int oi = s_idx[i];
        const float* gsrc = bx + (size_t)oi * 4;             // 16B aligned
        const unsigned ldsoff = (unsigned)(size_t)(&s_box[i][0]);
        asm volatile("global_load_async_to_lds_b128 %0, %1, off"
                     :: "v"(ldsoff), "v"(gsrc) : "memory");
    }
    asm volatile("s_wait_asynccnt 0x0" ::: "memory");
    __syncthreads();

    // greedy suppression over the valid prefix
    for (int i = 0; i < nvalid; ++i) {
        if (!s_sup[i]) {                         // workgroup-uniform branch
            const float ax1 = s_box[i][0], ay1 = s_box[i][1];
            const float ax2 = s_box[i][2], ay2 = s_box[i][3];
            const float aarea = (ax2 - ax1) * (ay2 - ay1);
            for (int j = i + 1 + tid; j < nvalid; j += 256) {
                if (!s_sup[j]) {
                    const float bx1 = s_box[j][0], by1 = s_box[j][1];
                    const float bx2 = s_box[j][2], by2 = s_box[j][3];
                    const float ix1 = fmaxf(ax1, bx1), iy1 = fmaxf(ay1, by1);
                    const float ix2 = fminf(ax2, bx2), iy2 = fminf(ay2, by2);
                    const float iw = fmaxf(ix2 - ix1, 0.0f);
                    const float ih = fmaxf(iy2 - iy1, 0.0f);
                    const float inter = iw * ih;
                    const float barea = (bx2 - bx1) * (by2 - by1);
                    const float iou = inter / (aarea + barea - inter + 1e-9f);
                    if (iou > NMS_THRESH) s_sup[j] = 1;
                }
            }
            __syncthreads();
        }
    }

    // scatter keep-masked scores back in original index order
    for (int i = tid; i < 1024; i += 256) {
        const int oi = s_idx[i];
        if (oi < NN) {
            const bool kept = (i < nvalid) && !s_sup[i];
            kept_scores[(size_t)pair * NN + oi] = kept ? s_key[i] : 0.0f;
        }
    }
}

// ---------------------------------------------------------------------------
// Kernel 4: per-image top-100 (tie-break lowest flat index, like top_k) and
// output assembly.  One 256-thread block per image.  Mutates kept_scores
// (fully rewritten by nms_kernel every launch -> replay-safe).
// ---------------------------------------------------------------------------
__global__ __launch_bounds__(256)
void topk_kernel(float* __restrict__ kept_scores, const float* __restrict__ boxes,
                 float* __restrict__ out)
{
    __shared__ float r_val[256];
    __shared__ int   r_idx[256];
    const int tid = threadIdx.x;
    const int b = blockIdx.x;
    float* sc = kept_scores + (size_t)b * NCLS * NN;       // 80000 entries
    const int M = NCLS * NN;

    for (int kk = 0; kk < DETS_PER_IMG; ++kk) {
        float v = -3.0f; int id = M;
        for (int j = tid; j < M; j += 256) {
            const float s = sc[j];
            if (s > v) { v = s; id = j; }                  // ascending scan -> lowest idx on tie
        }
        r_val[tid] = v; r_idx[tid] = id;
        __syncthreads();
        for (int off = 128; off > 0; off >>= 1) {
            if (tid < off) {
                const float ov = r_val[tid + off]; const int oi = r_idx[tid + off];
                if (ov > r_val[tid] || (ov == r_val[tid] && oi < r_idx[tid])) {
                    r_val[tid] = ov; r_idx[tid] = oi;
                }
            }
            __syncthreads();
        }
        if (tid == 0) {
            const float topv = r_val[0]; const int topi = r_idx[0];
            const bool vt = topv > 0.0f;
            const float* bp = boxes + ((size_t)b * M + topi) * 4;
            float* d = out + (size_t)b * DETS_PER_IMG * 5 + kk * 5;
            d[0] = vt ? bp[0] : 0.0f;
            d[1] = vt ? bp[1] : 0.0f;
            d[2] = vt ? bp[2] : 0.0f;
            d[3] = vt ? bp[3] : 0.0f;
            d[4] = topv;
            out[BB * DETS_PER_IMG * 5 + (size_t)b * DETS_PER_IMG + kk] =
                vt ? (float)(1 + topi / NN) : -1.0f;       // label (as float)
            sc[topi] = -1.0f;                              // exclude from next pass
        }
        __syncthreads();
    }
}

// ---------------------------------------------------------------------------
extern "C" void kernel_launch(void* const* d_in, const int* in_sizes, int n_in,
                              void* d_out, int out_size, void* d_ws, size_t ws_size,
                              hipStream_t stream) {
    (void)in_sizes; (void)n_in; (void)out_size; (void)ws_size;
    const float* class_logits   = (const float*)d_in[0];   // [8000, 81]
    const float* box_regression = (const float*)d_in[1];   // [8000, 324]
    const float* proposals      = (const float*)d_in[2];   // [8000, 4]
    // d_in[3] = features: unused by reference.
    float* out = (float*)d_out;                            // dets(4000) ++ labels(800)

    float* ws     = (float*)d_ws;
    float* scores = ws;                       // [8][80][1000]        640000 f
    float* boxes  = ws + 640000;              // [8][80][1000][4]    2560000 f

    softmax_kernel<<<125, 128, 0, stream>>>(class_logits, scores);
    decode_kernel<<<(BB * NN * NCLS + 255) / 256, 256, 0, stream>>>(
        box_regression, proposals, boxes);
    nms_kernel<<<BB * NCLS, 256, 0, stream>>>(scores, boxes, scores);
    topk_kernel<<<BB, 256, 0, stream>>>(scores, boxes, out);
}